// ToeplitzMemoryProjection_52956946760026
// MI455X (gfx1250) — compile-verified
//
#include <hip/hip_runtime.h>
#include <hip/hip_bf16.h>

typedef float v2f __attribute__((ext_vector_type(2)));
typedef float v8f __attribute__((ext_vector_type(8)));

#define L_SEQ 1024
#define BM    256   // B*M = 4*64 independent sequences, and (L,B,M) flat stride
#define NORD  256   // HiPPO order N

// One workgroup per (b,m): 128 threads = 4 wave32.
// State x (256 f32) lives in LDS; per step l we
//  1) build As kernel + Bs vector from closed-form geometric series,
//  2) do the block-Toeplitz matvec Y = sum_d T_d * shift_d(X) with
//     V_WMMA_F32_16X16X4_F32 (full fp32), 4 diagonals per wave,
//  3) reduce the 4 partial C tiles + Bs in LDS, update x, store to out.
__launch_bounds__(128)
__global__ void hippo_toeplitz_scan(const float* __restrict__ inp,
                                    const float* __restrict__ dt,
                                    float* __restrict__ out)
{
    __shared__ __align__(32) float a_lds[NORD];      // As kernel (first Toeplitz column)
    __shared__ __align__(32) float b_lds[NORD];      // Bs vector
    __shared__ __align__(32) float x_lds[NORD];      // running state
    __shared__ __align__(32) float part[4][NORD];    // per-wave partial results

    const int bm   = blockIdx.x;
    const int tid  = threadIdx.x;
    const int w    = tid >> 5;      // wave id 0..3
    const int lane = tid & 31;
    const int hi   = lane >> 4;     // half-wave select
    const int lo   = lane & 15;     // A: row M; B/C: column N

    x_lds[tid]       = 0.0f;
    x_lds[tid + 128] = 0.0f;
    __syncthreads();

    for (int l = 0; l < L_SEQ; ++l) {
        const int   sidx = l * BM + bm;           // (L,B,M) flat index
        const float dtv  = dt[sidx];              // uniform -> scalar load
        const float u    = inp[sidx];
        const float del  = 0.5f * dtv;            // alpha*dt = (1-alpha)*dt
        const float c    = 1.0f + 0.5f * del;     // 1 + del*(1+a), a = -0.5
        const float r    = del / c;
        const float omr  = 1.0f - r;              // = As[0]
        const float g    = __logf(omr);
        const float sa   = 2.0f * del / (c * c);  // |As[n>=1]| scale
        const float sb   = dtv * u / c;           // Bs scale

        // Phase 1: As[n] and Bs[n] as geometric series (no FFT needed).
        #pragma unroll
        for (int rep = 0; rep < 2; ++rep) {
            const int   n    = tid + rep * 128;
            const float pnm1 = __expf((float)(n - 1) * g);  // (1-r)^(n-1)
            const float pn   = pnm1 * omr;                  // (1-r)^n
            b_lds[n] = sb * pn;
            a_lds[n] = (n == 0) ? omr : (-sa * pnm1);
        }
        __syncthreads();

        // Phase 2: Y += T_d * shift_d(X) for this wave's diagonals d = w+4*dd.
        v8f acc0 = {0.f, 0.f, 0.f, 0.f, 0.f, 0.f, 0.f, 0.f};
        v8f acc1 = acc0;
        #pragma unroll
        for (int dd = 0; dd < 4; ++dd) {
            const int  d     = w + 4 * dd;          // block diagonal 0..15
            const bool bval  = (lo >= d);           // column j=lo gets shift-d term only if j>=d
            const int  xbase = 16 * (lo - d);
            #pragma unroll
            for (int t = 0; t < 4; ++t) {           // K split into 4 chunks of 4
                const int kb  = 4 * t + 2 * hi;     // per-lane K base (A/B fragment layout)
                const int ia0 = 16 * d + lo - kb;   // Toeplitz: T_d[p,q] = a[16d+p-q]
                const int ia1 = ia0 - 1;
                float a0 = a_lds[ia0 < 0 ? 0 : ia0];
                float a1 = a_lds[ia1 < 0 ? 0 : ia1];
                a0 = (ia0 >= 0) ? a0 : 0.0f;        // branchless zero outside band (EXEC stays full)
                a1 = (ia1 >= 0) ? a1 : 0.0f;
                float b0 = x_lds[bval ? (xbase + kb)     : 0];
                float b1 = x_lds[bval ? (xbase + kb + 1) : 0];
                b0 = bval ? b0 : 0.0f;
                b1 = bval ? b1 : 0.0f;
                const v2f av = {a0, a1};
                const v2f bv = {b0, b1};
                // 8 args: (neg_a, A, neg_b, B, c_mod, C, reuse_a, reuse_b)
                if (dd & 1)
                    acc1 = __builtin_amdgcn_wmma_f32_16x16x4_f32(
                        false, av, false, bv, (short)0, acc1, false, false);
                else
                    acc0 = __builtin_amdgcn_wmma_f32_16x16x4_f32(
                        false, av, false, bv, (short)0, acc0, false, false);
            }
        }
        const v8f ysum = acc0 + acc1;
        // C layout: VGPR g -> element y[16*lo + 8*hi + g]; 32B-aligned contiguous store.
        *(v8f*)&part[w][16 * lo + 8 * hi] = ysum;
        __syncthreads();

        // Phase 3: reduce 4 waves' partials, add Bs, update state, store output.
        #pragma unroll
        for (int rep = 0; rep < 2; ++rep) {
            const int   n = tid + rep * 128;
            const float y = part[0][n] + part[1][n] + part[2][n] + part[3][n] + b_lds[n];
            x_lds[n] = y;
            out[(size_t)sidx * NORD + n] = y;
        }
        __syncthreads();
    }
}

extern "C" void kernel_launch(void* const* d_in, const int* in_sizes, int n_in,
                              void* d_out, int out_size, void* d_ws, size_t ws_size,
                              hipStream_t stream) {
    (void)in_sizes; (void)n_in; (void)out_size; (void)d_ws; (void)ws_size;
    const float* inp = (const float*)d_in[0];   // inputs (L,B,M) f32
    const float* dtp = (const float*)d_in[1];   // dt     (L,B,M) f32
    float*       out = (float*)d_out;           // out    (L,B,M,N) f32
    hippo_toeplitz_scan<<<dim3(BM), dim3(128), 0, stream>>>(inp, dtp, out);
}